// ContentAttention_66812511256673
// MI455X (gfx1250) — compile-verified
//
#include <hip/hip_runtime.h>
#include <math.h>

typedef __attribute__((ext_vector_type(2))) float v2f;
typedef __attribute__((ext_vector_type(8))) float v8f;

#define B_   32
#define Q_   512
#define K_   512
#define D_   1024
#define TILE 128      // output tile per workgroup (128x128)
#define KC   32       // k-slab staged in LDS per iteration
#define LSTR 36       // padded LDS row stride (floats): 144B rows, conflict-free, 16B-aligned
#define NSLAB (D_ / KC)

// Async global->LDS copy (GLOBAL_LOAD_ASYNC_TO_LDS_B128, ASYNCcnt-tracked).
// lds_off: 32-bit LDS byte offset (generic LDS addr truncates to offset, ISA 10.2)
// gptr   : 64-bit global address (VGPR pair)
__device__ __forceinline__ void async_ld_b128(unsigned lds_off, const void* gptr) {
    asm volatile("global_load_async_to_lds_b128 %0, %1, off"
                 :
                 : "v"(lds_off), "v"(gptr)
                 : "memory");
}
__device__ __forceinline__ void wait_async0() {
    asm volatile("s_wait_asynccnt 0x0" ::: "memory");
}

// ---------------------------------------------------------------------------
// Kernel 1: per-batch logits = Q * K^T with fp32 WMMA; double-buffered async
// global->LDS pipeline; mask applied; unnormalized logits to attn region.
// Block = 256 threads = 8 wave32; wave (wm in 0..3, wn in 0..1) computes a
// 32x64 sub-tile as 2x4 WMMA 16x16 fp32 accumulators.
// ---------------------------------------------------------------------------
__global__ __launch_bounds__(256) void ca_gemm_logits(
    const float* __restrict__ qg, const float* __restrict__ kg,
    float* __restrict__ out)
{
    __shared__ float Qs[2][TILE * LSTR];
    __shared__ float Ks[2][TILE * LSTR];

    const int b  = blockIdx.z;
    const int q0 = blockIdx.y * TILE;
    const int k0 = blockIdx.x * TILE;
    const float* __restrict__ qb = qg + (size_t)b * Q_ * D_;
    const float* __restrict__ kb = kg + (size_t)b * K_ * D_;

    const int tid   = threadIdx.x;
    const int wave  = tid >> 5;
    const int lane  = tid & 31;
    const int half  = lane >> 4;     // wave32: lanes 16-31 hold K+2..K+3 / M+8
    const int lr    = lane & 15;
    const int mbase = (wave >> 1) * 32;   // 0,32,64,96
    const int nbase = (wave & 1) * 64;    // 0,64

    // Per-thread staging assignment: 4 float4 per matrix per slab.
    const int f0_row[4] = { (tid + 0*256) >> 3, (tid + 1*256) >> 3,
                            (tid + 2*256) >> 3, (tid + 3*256) >> 3 };
    const int f0_cv [4] = { ((tid + 0*256) & 7) * 4, ((tid + 1*256) & 7) * 4,
                            ((tid + 2*256) & 7) * 4, ((tid + 3*256) & 7) * 4 };

    auto issue_slab = [&](int buf, int kslab) {
        #pragma unroll
        for (int j = 0; j < 4; ++j) {
            const int row = f0_row[j];
            const int cv  = f0_cv[j];
            const unsigned qoff = (unsigned)(uintptr_t)(&Qs[buf][row * LSTR + cv]);
            const unsigned koff = (unsigned)(uintptr_t)(&Ks[buf][row * LSTR + cv]);
            async_ld_b128(qoff, qb + (size_t)(q0 + row) * D_ + kslab + cv);
            async_ld_b128(koff, kb + (size_t)(k0 + row) * D_ + kslab + cv);
        }
    };

    v8f acc[2][4] = {};

    issue_slab(0, 0);   // prologue: slab 0 -> buffer 0

    for (int s = 0; s < NSLAB; ++s) {
        const int cur = s & 1;
        wait_async0();      // this wave's copies into buf[cur] have landed
        __syncthreads();    // everyone's copies landed; everyone done reading buf[cur^1]
        if (s + 1 < NSLAB) issue_slab(cur ^ 1, (s + 1) * KC);

        const float* __restrict__ Qt = Qs[cur];
        const float* __restrict__ Kt = Ks[cur];

        // 8 k-steps of 4; per step: 2 A frags, 4 B frags, 8 WMMAs.
        #pragma unroll
        for (int kk = 0; kk < KC; kk += 4) {
            const int koff = kk + half * 2;   // lanes 0-15: K=kk,kk+1; 16-31: kk+2,kk+3
            v2f afr[2], bfr[4];
            #pragma unroll
            for (int mi = 0; mi < 2; ++mi)
                afr[mi] = *(const v2f*)(&Qt[(mbase + mi * 16 + lr) * LSTR + koff]);
            #pragma unroll
            for (int ni = 0; ni < 4; ++ni)
                bfr[ni] = *(const v2f*)(&Kt[(nbase + ni * 16 + lr) * LSTR + koff]);
            #pragma unroll
            for (int mi = 0; mi < 2; ++mi)
                #pragma unroll
                for (int ni = 0; ni < 4; ++ni)
                    acc[mi][ni] = __builtin_amdgcn_wmma_f32_16x16x4_f32(
                        false, afr[mi], false, bfr[ni],
                        (short)0, acc[mi][ni], false, false);
        }
    }

    // Epilogue: C/D layout -> lane lr = N, VGPR r = M row (M = r + 8*half).
    const float NEG_INF = -__builtin_inff();
    #pragma unroll
    for (int ni = 0; ni < 4; ++ni) {
        const int col = k0 + nbase + ni * 16 + lr;
        const bool masked = (kb[(size_t)col * D_] == 0.0f);  // keys[b,col,0]==0
        #pragma unroll
        for (int mi = 0; mi < 2; ++mi) {
            const int qrow0 = q0 + mbase + mi * 16 + half * 8;
            float* orow = out + ((size_t)b * Q_ + qrow0) * K_ + col;
            #pragma unroll
            for (int r = 0; r < 8; ++r) {
                const float v = acc[mi][ni][r];
                orow[(size_t)r * K_] = masked ? NEG_INF : v;
            }
        }
    }
}

// ---------------------------------------------------------------------------
// Kernel 2: one wave32 per (b,q) row of 512 logits. Coalesced float4 reads,
// shfl_xor reductions for max and sum(exp), in-place normalized write,
// lane 0 writes confidence = sigmoid((m + log(sum) + bias) * temperature).
// ---------------------------------------------------------------------------
__global__ __launch_bounds__(256) void ca_softmax_conf(
    float* __restrict__ attn, float* __restrict__ conf,
    const float* __restrict__ temp_p, const float* __restrict__ bias_p)
{
    const int wave = threadIdx.x >> 5;
    const int lane = threadIdx.x & 31;
    const int row  = blockIdx.x * 8 + wave;    // 0 .. B*Q-1
    float* rp = attn + (size_t)row * K_;

    float4 v[4];
    #pragma unroll
    for (int j = 0; j < 4; ++j) v[j] = ((const float4*)rp)[lane + j * 32];

    float m = -__builtin_inff();
    #pragma unroll
    for (int j = 0; j < 4; ++j)
        m = fmaxf(m, fmaxf(fmaxf(v[j].x, v[j].y), fmaxf(v[j].z, v[j].w)));
    #pragma unroll
    for (int off = 16; off > 0; off >>= 1) m = fmaxf(m, __shfl_xor(m, off, 32));

    float s = 0.0f;
    #pragma unroll
    for (int j = 0; j < 4; ++j) {
        v[j].x = expf(v[j].x - m); v[j].y = expf(v[j].y - m);
        v[j].z = expf(v[j].z - m); v[j].w = expf(v[j].w - m);
        s += (v[j].x + v[j].y) + (v[j].z + v[j].w);
    }
    #pragma unroll
    for (int off = 16; off > 0; off >>= 1) s += __shfl_xor(s, off, 32);

    const float inv = 1.0f / s;
    #pragma unroll
    for (int j = 0; j < 4; ++j) {
        v[j].x *= inv; v[j].y *= inv; v[j].z *= inv; v[j].w *= inv;
        ((float4*)rp)[lane + j * 32] = v[j];
    }

    if (lane == 0) {
        const float lse = m + logf(s);
        const float z   = (lse + bias_p[0]) * temp_p[0];
        conf[row] = 1.0f / (1.0f + expf(-z));
    }
}

// ---------------------------------------------------------------------------
extern "C" void kernel_launch(void* const* d_in, const int* in_sizes, int n_in,
                              void* d_out, int out_size, void* d_ws, size_t ws_size,
                              hipStream_t stream)
{
    (void)in_sizes; (void)n_in; (void)out_size; (void)d_ws; (void)ws_size;
    const float* queries     = (const float*)d_in[0];
    const float* keys        = (const float*)d_in[1];
    const float* temperature = (const float*)d_in[2];
    const float* bias        = (const float*)d_in[3];

    float* attn = (float*)d_out;                       // (B, Q, K)
    float* conf = attn + (size_t)B_ * Q_ * K_;         // (B, Q)

    dim3 grid1(K_ / TILE, Q_ / TILE, B_);              // (4, 4, 32)
    ca_gemm_logits<<<grid1, 256, 0, stream>>>(queries, keys, attn);

    ca_softmax_conf<<<(B_ * Q_) / 8, 256, 0, stream>>>(attn, conf, temperature, bias);
}